// Decoder_14405320311384
// MI455X (gfx1250) — compile-verified
//
#include <hip/hip_runtime.h>
#include <hip/hip_bf16.h>
#include <math.h>

#define B 32
#define E 256
#define H 512
#define V 32000
#define T 128
#define HS_STRIDE 516  // pad 512-float rows to dodge LDS bank conflicts

typedef __attribute__((ext_vector_type(2))) float v2f;
typedef __attribute__((ext_vector_type(8))) float v8f;

__device__ __forceinline__ float sigmoid_f(float x) {
    return 1.0f / (1.0f + expf(-x));
}

// ---------------------------------------------------------------------------
// init: h0 <- context_vector, tok <- START_IDX (=1)
// ---------------------------------------------------------------------------
__global__ __launch_bounds__(256) void init_state(const float* __restrict__ ctx,
                                                  float* __restrict__ h0,
                                                  int* __restrict__ tok) {
    int idx = blockIdx.x * 256 + threadIdx.x;
    if (idx < B * H) h0[idx] = ctx[idx];
    if (idx < B) tok[idx] = 1;
}

// ---------------------------------------------------------------------------
// GRU cell: one thread per (b, j) output element. Embedding gather fused.
// grid = (B*H)/256 = 64 blocks, 256 threads. Lanes within a wave share j.
// ---------------------------------------------------------------------------
__global__ __launch_bounds__(256) void gru_step(const float* __restrict__ embed,
                                                const float* __restrict__ w_ih,
                                                const float* __restrict__ w_hh,
                                                const float* __restrict__ b_ih,
                                                const float* __restrict__ b_hh,
                                                const float* __restrict__ h_in,
                                                float* __restrict__ h_out,
                                                const int* __restrict__ tok) {
    int tid = blockIdx.x * 256 + threadIdx.x;   // 0..B*H-1
    int b = tid & (B - 1);
    int j = tid >> 5;                           // 0..H-1

    // input gates: x = embed_table[tok[b]], rows j, j+H, j+2H of w_ih
    const float* x  = embed + (size_t)tok[b] * E;
    const float* wr = w_ih + (size_t)j * E;
    const float* wz = w_ih + (size_t)(j + H) * E;
    const float* wn = w_ih + (size_t)(j + 2 * H) * E;
    float ir = b_ih[j], iz = b_ih[j + H], inn = b_ih[j + 2 * H];
    #pragma unroll 4
    for (int k = 0; k < E; ++k) {
        float xv = x[k];
        ir += xv * wr[k];
        iz += xv * wz[k];
        inn += xv * wn[k];
    }

    // hidden gates
    const float* hrow = h_in + (size_t)b * H;
    const float* ur = w_hh + (size_t)j * H;
    const float* uz = w_hh + (size_t)(j + H) * H;
    const float* un = w_hh + (size_t)(j + 2 * H) * H;
    float hr = b_hh[j], hz = b_hh[j + H], hn = b_hh[j + 2 * H];
    #pragma unroll 4
    for (int k = 0; k < H; ++k) {
        float hv = hrow[k];
        hr += hv * ur[k];
        hz += hv * uz[k];
        hn += hv * un[k];
    }

    float r = sigmoid_f(ir + hr);
    float z = sigmoid_f(iz + hz);
    float n = tanhf(inn + r * hn);
    h_out[(size_t)b * H + j] = (1.0f - z) * n + z * hrow[j];
}

// ---------------------------------------------------------------------------
// logits = h @ fc_w.T + fc_b via V_WMMA_F32_16X16X4_F32.
// grid = V/128 = 250 blocks, 8 waves. Wave w owns the 16-wide V tile
// [blk*128 + w*16, +16) and both M-tiles (M=32 -> two 16x16 accumulators).
// h staged in LDS once per block; fc_w streams from L2 (fits: 65.5 MB < 192 MB).
//
// WMMA 16x16x4 f32 operand layout (ISA 7.12.2):
//   A (16x4): lane l holds M=l&15, K = (l>>4)*2 + {0,1}  -> v2f
//   B (4x16): lane l holds N=l&15, K = (l>>4)*2 + {0,1}  -> v2f
//   C (16x16): vgpr i, lane l -> M = i + (l>>4)*8, N = l&15
// ---------------------------------------------------------------------------
__global__ __launch_bounds__(256) void logits_wmma(const float* __restrict__ fc_w,
                                                   const float* __restrict__ fc_b,
                                                   const float* __restrict__ h,
                                                   float* __restrict__ out,
                                                   int t) {
    __shared__ float hs[B * HS_STRIDE];

    // stage h[32][512] into LDS with padded stride
    for (int i = threadIdx.x; i < B * H; i += 256) {
        int bb = i >> 9;          // /512
        int kk = i & (H - 1);
        hs[bb * HS_STRIDE + kk] = h[i];
    }
    __syncthreads();

    int lane = threadIdx.x & 31;
    int wave = threadIdx.x >> 5;          // 0..7
    int n = lane & 15;                    // N (and A's M) index within tile
    int khalf = (lane >> 4) * 2;          // K sub-offset per half-wave
    int vcol = blockIdx.x * 128 + wave * 16 + n;

    const float* fwrow = fc_w + (size_t)vcol * H;
    const float* a0row = hs + n * HS_STRIDE;
    const float* a1row = hs + (n + 16) * HS_STRIDE;

    v8f c0 = {};
    v8f c1 = {};
    for (int k = 0; k < H; k += 4) {
        v2f a0 = { a0row[k + khalf], a0row[k + khalf + 1] };   // rows 0..15
        v2f a1 = { a1row[k + khalf], a1row[k + khalf + 1] };   // rows 16..31
        v2f bv = { fwrow[k + khalf], fwrow[k + khalf + 1] };
        c0 = __builtin_amdgcn_wmma_f32_16x16x4_f32(false, a0, false, bv,
                                                   (short)0, c0, false, false);
        c1 = __builtin_amdgcn_wmma_f32_16x16x4_f32(false, a1, false, bv,
                                                   (short)0, c1, false, false);
    }

    float bias = fc_b[vcol];
    size_t colbase = (size_t)t * V + vcol;
    #pragma unroll
    for (int i = 0; i < 8; ++i) {
        int m0 = i + (lane >> 4) * 8;     // rows 0..15 across the two half-waves
        out[(size_t)m0 * ((size_t)T * V) + colbase] = c0[i] + bias;
        out[(size_t)(m0 + 16) * ((size_t)T * V) + colbase] = c1[i] + bias;
    }
}

// ---------------------------------------------------------------------------
// argmax over V per batch row (first-index tie-break, matching jnp.argmax),
// writes greedy token feedback + prediction (as float) into d_out tail.
// ---------------------------------------------------------------------------
__global__ __launch_bounds__(256) void argmax_step(const float* __restrict__ out,
                                                   int t,
                                                   int* __restrict__ tok,
                                                   float* __restrict__ preds) {
    __shared__ float sv[256];
    __shared__ int si[256];
    int b = blockIdx.x;
    int tid = threadIdx.x;
    const float* row = out + (size_t)b * ((size_t)T * V) + (size_t)t * V;

    float best = -INFINITY;
    int bi = V;
    for (int v = tid; v < V; v += 256) {
        float x = row[v];
        if (x > best) { best = x; bi = v; }   // increasing v => keeps first max
    }
    sv[tid] = best;
    si[tid] = bi;
    __syncthreads();
    for (int s = 128; s > 0; s >>= 1) {
        if (tid < s) {
            float ov = sv[tid + s];
            int oi = si[tid + s];
            if (ov > sv[tid] || (ov == sv[tid] && oi < si[tid])) {
                sv[tid] = ov;
                si[tid] = oi;
            }
        }
        __syncthreads();
    }
    if (tid == 0) {
        tok[b] = si[0];
        preds[(size_t)b * T + t] = (float)si[0];
    }
}

// ---------------------------------------------------------------------------
extern "C" void kernel_launch(void* const* d_in, const int* in_sizes, int n_in,
                              void* d_out, int out_size, void* d_ws, size_t ws_size,
                              hipStream_t stream) {
    const float* ctx   = (const float*)d_in[0];   // [B, H]
    const float* embed = (const float*)d_in[1];   // [V, E]
    const float* w_ih  = (const float*)d_in[2];   // [3H, E]
    const float* w_hh  = (const float*)d_in[3];   // [3H, H]
    const float* b_ih  = (const float*)d_in[4];   // [3H]
    const float* b_hh  = (const float*)d_in[5];   // [3H]
    const float* fc_w  = (const float*)d_in[6];   // [V, H]
    const float* fc_b  = (const float*)d_in[7];   // [V]

    float* out   = (float*)d_out;                       // [B, T, V] logits
    float* preds = out + (size_t)B * T * V;             // [B, T] predictions

    float* hbuf0 = (float*)d_ws;
    float* hbuf1 = hbuf0 + B * H;
    int*   tok   = (int*)(hbuf1 + B * H);

    init_state<<<64, 256, 0, stream>>>(ctx, hbuf0, tok);

    int cur = 0;
    for (int t = 0; t < T; ++t) {
        float* h_in  = cur ? hbuf1 : hbuf0;
        float* h_out = cur ? hbuf0 : hbuf1;
        gru_step<<<(B * H) / 256, 256, 0, stream>>>(embed, w_ih, w_hh, b_ih, b_hh,
                                                    h_in, h_out, tok);
        logits_wmma<<<V / 128, 256, 0, stream>>>(fc_w, fc_b, h_out, out, t);
        argmax_step<<<B, 256, 0, stream>>>(out, t, tok, preds);
        cur ^= 1;
    }
}